// MaxUnpooling2D_91044716741200
// MI455X (gfx1250) — compile-verified
//
#include <hip/hip_runtime.h>

// MaxUnpooling2D scatter-add for MI455X (gfx1250, wave32).
//
// dst = b*Ho*Wo*C + (idx & ~(C-1)) + (elem % C), accumulate with f32 atomics.
// Memory-bound: ~128 MiB NT reads + 256 MiB zero stores + random 4B atomic adds
// merged in the 192 MB L2. No matrix math -> no WMMA; the CDNA5 paths used are
// B128 NT loads, B128 stores, and no-return global_atomic_add_f32.

typedef float v4f __attribute__((ext_vector_type(4)));
typedef int   v4i __attribute__((ext_vector_type(4)));

namespace {
constexpr int kB   = 16;
constexpr int kH   = 128;
constexpr int kW   = 128;
constexpr int kC   = 64;                 // power of two
constexpr int kHo  = kH * 2;
constexpr int kWo  = kW * 2;
constexpr int kHWC   = kH * kW * kC;     // 1 << 20 (input elems per batch)
constexpr int kHoWoC = kHo * kWo * kC;   // 1 << 22 (output elems per batch)
constexpr int kInElems  = kB * kHWC;     // 1 << 24
constexpr int kOutElems = kB * kHoWoC;   // 1 << 26
constexpr int kThreads  = 256;           // 8 wave32 waves
}

// Pass 1: zero the 256 MiB output with 16-byte stores (exact grid, no tail).
__global__ __launch_bounds__(kThreads) void unpool_zero_f4(v4f* __restrict__ out4) {
    const int i = blockIdx.x * kThreads + threadIdx.x;
    v4f z = {0.0f, 0.0f, 0.0f, 0.0f};
    out4[i] = z;
}

// Pass 2: each thread handles 4 consecutive elements (one B128 feature load +
// one B128 index load), then issues 4 fire-and-forget global_atomic_add_f32.
__global__ __launch_bounds__(kThreads) void unpool_scatter_add4(
    const v4f* __restrict__ feat4,
    const v4i* __restrict__ idx4,
    float*     __restrict__ out) {
    const int t = blockIdx.x * kThreads + threadIdx.x;   // 0 .. 2^22 - 1
    const int e = t << 2;                                // flat element base

    // Streamed-once inputs: non-temporal so they don't evict output lines
    // from L2 (output working set 256 MiB vs 192 MB L2).
    const v4f v  = __builtin_nontemporal_load(feat4 + t);
    const v4i id = __builtin_nontemporal_load(idx4 + t);

    const int base = (e >> 20) << 22;        // batch * Ho*Wo*C
    const int c0   = e & (kC - 1);           // channels c0 .. c0+3 (c0 % 4 == 0)
    float* __restrict__ p = out + base + c0;

    constexpr int kRowMask = ~(kC - 1);
    // Discarded return -> no-return atomic (STOREcnt, no wait-for-old-value).
    unsafeAtomicAdd(p + ((id.x & kRowMask) + 0), v.x);
    unsafeAtomicAdd(p + ((id.y & kRowMask) + 1), v.y);
    unsafeAtomicAdd(p + ((id.z & kRowMask) + 2), v.z);
    unsafeAtomicAdd(p + ((id.w & kRowMask) + 3), v.w);
}

extern "C" void kernel_launch(void* const* d_in, const int* in_sizes, int n_in,
                              void* d_out, int out_size, void* d_ws, size_t ws_size,
                              hipStream_t stream) {
    const v4f* feat4 = (const v4f*)d_in[0];   // float32 features, 16*128*128*64
    const v4i* idx4  = (const v4i*)d_in[1];   // int32 indices, same shape
    float*     out   = (float*)d_out;         // float32, 16*256*256*64
    (void)in_sizes; (void)n_in; (void)out_size; (void)d_ws; (void)ws_size;

    // Pass 1: zero-fill. kOutElems/4 vec4 stores, 256 threads/block -> 65536 blocks.
    {
        const int n4     = kOutElems / 4;
        const int blocks = n4 / kThreads;
        unpool_zero_f4<<<blocks, kThreads, 0, stream>>>((v4f*)out);
    }

    // Pass 2: scatter-add. kInElems/4 threads -> 16384 blocks.
    {
        const int nt     = kInElems / 4;
        const int blocks = nt / kThreads;
        unpool_scatter_add4<<<blocks, kThreads, 0, stream>>>(feat4, idx4, out);
    }
}